// DGISubgraphCL_24292335026459
// MI455X (gfx1250) — compile-verified
//
#include <hip/hip_runtime.h>

typedef float v2f __attribute__((ext_vector_type(2)));
typedef float v8f __attribute__((ext_vector_type(8)));

#define TPB 256
#define MAX_BFS_ITERS 32
#define BS_ITERS 17
#define WSTRIDE 66  // LDS row stride for transposed W (bank-conflict-free)

// ---------------- small math ----------------
__device__ __forceinline__ float softplusf(float x) {
  // stable: max(x,0) + log1p(exp(-|x|))   (matches jax.nn.softplus)
  return fmaxf(x, 0.0f) + log1pf(expf(-fabsf(x)));
}

// ---------------- degrees ----------------
__global__ void k_deg(const int* __restrict__ dst, float* __restrict__ deg, int nE) {
  int e = blockIdx.x * blockDim.x + threadIdx.x;
  if (e < nE) atomicAdd(&deg[dst[e]], 1.0f);
}

// ---------------- feature aggregation: one wave per edge ----------------
__global__ void k_agg(const float* __restrict__ feat, const int* __restrict__ src,
                      const int* __restrict__ dst, const int* __restrict__ perm,
                      float* __restrict__ agg, int nE) {
  int t = blockIdx.x * blockDim.x + threadIdx.x;
  int e = t >> 5, lane = t & 31;
  if (e >= nE) return;
  int s = src[e];
  if (perm) s = perm[s];
  int d = dst[e];
  const float* xs = feat + (size_t)s * 64u;
  float* ad = agg + (size_t)d * 64u;
  atomicAdd(&ad[lane], xs[lane]);
  atomicAdd(&ad[lane + 32], xs[lane + 32]);
}

__global__ void k_agg_masked(const float* __restrict__ feat, const int* __restrict__ src,
                             const int* __restrict__ dst, const int* __restrict__ keep,
                             float* __restrict__ agg, float* __restrict__ degm, int nE) {
  int t = blockIdx.x * blockDim.x + threadIdx.x;
  int e = t >> 5, lane = t & 31;
  if (e >= nE) return;
  int s = src[e], d = dst[e];
  if (!(keep[s] && keep[d])) return;  // edge index is wave-uniform
  const float* xs = feat + (size_t)s * 64u;
  float* ad = agg + (size_t)d * 64u;
  atomicAdd(&ad[lane], xs[lane]);
  atomicAdd(&ad[lane + 32], xs[lane + 32]);
  if (lane == 0) atomicAdd(&degm[d], 1.0f);
}

// ---------------- WMMA GEMM tile: one wave -> rows [rowBase,rowBase+16) x 64 cols ----
// A (16x4 f32): lanes 0-15 hold (M=lane, K=k0..k0+1) in v2f; lanes 16-31 K=k0+2..k0+3.
// B (4x16 f32): VGPR0 lanes 0-15 = row k0, lanes 16-31 = row k0+2; VGPR1 rows k0+1/k0+3.
// C/D (16x16):  VGPR r: lanes 0-15 -> (M=r, N=lane); lanes 16-31 -> (M=r+8, N=lane-16).
// W is staged in LDS *transposed* with row stride WSTRIDE so each lane's B fragment
// {W[ka][col], W[ka+1][col]} = sWt[col*WSTRIDE + ka..ka+1] is one aligned ds_load_b64.
#if defined(__gfx1250__) && __has_builtin(__builtin_amdgcn_wmma_f32_16x16x4_f32)
#define HAVE_WMMA_F32 1
#else
#define HAVE_WMMA_F32 0
#endif

__device__ __forceinline__ void load_Wt(const float* __restrict__ W, float* __restrict__ sWt) {
  for (int i = threadIdx.x; i < 4096; i += blockDim.x) {
    int k = i >> 6, c = i & 63;
    sWt[c * WSTRIDE + k] = W[i];  // coalesced global read, one-time scatter to LDS
  }
}

__device__ __forceinline__ void gemm_tile(const float* __restrict__ agg,
                                          const float* __restrict__ deg,
                                          const float* __restrict__ sWt,  // 64xWSTRIDE LDS
                                          int rowBase, int nRows, v8f acc[4]) {
  const int lane = threadIdx.x & 31;
  const int half = lane >> 4;
  const int m = lane & 15;
  int row = rowBase + m;
  if (row >= nRows) row = nRows - 1;  // clamp keeps EXEC all-ones
#if HAVE_WMMA_F32
  const float inv = 1.0f / fmaxf(deg[row], 1.0f);
  const float* ar = agg + (size_t)row * 64u;
#pragma unroll
  for (int k0 = 0; k0 < 64; k0 += 4) {
    const int ka = k0 + half * 2;
    v2f a = *(const v2f*)(ar + ka);  // global_load_b64
    a.x *= inv;
    a.y *= inv;
#pragma unroll
    for (int nt = 0; nt < 4; ++nt) {
      const int col = nt * 16 + m;
      v2f b = *(const v2f*)(sWt + col * WSTRIDE + ka);  // ds_load_b64, even-aligned
      acc[nt] = __builtin_amdgcn_wmma_f32_16x16x4_f32(
          /*neg_a=*/false, a, /*neg_b=*/false, b,
          /*c_mod=*/(short)0, acc[nt], /*reuse_a=*/false, /*reuse_b=*/false);
    }
  }
#else
  // scalar fallback producing the same per-lane layout
#pragma unroll
  for (int nt = 0; nt < 4; ++nt) {
    const int col = nt * 16 + m;
#pragma unroll
    for (int r = 0; r < 8; ++r) {
      int rr = rowBase + r + half * 8;
      if (rr >= nRows) rr = nRows - 1;
      const float inv = 1.0f / fmaxf(deg[rr], 1.0f);
      const float* ar = agg + (size_t)rr * 64u;
      float s = 0.0f;
      for (int k = 0; k < 64; ++k) s += ar[k] * sWt[col * WSTRIDE + k];
      acc[nt][r] = s * inv;
    }
  }
#endif
}

__global__ void __launch_bounds__(TPB) k_gemm_relu_store(
    const float* __restrict__ agg, const float* __restrict__ deg,
    const float* __restrict__ W, float* __restrict__ out, int nTiles, int nRows) {
  __shared__ float sWt[64 * WSTRIDE];
  load_Wt(W, sWt);
  __syncthreads();
  const int tile = blockIdx.x * (TPB / 32) + (threadIdx.x >> 5);
  if (tile >= nTiles) return;  // wave-uniform; no barriers after this
  v8f acc[4] = {};
  const int rowBase = tile * 16;
  gemm_tile(agg, deg, sWt, rowBase, nRows, acc);
  const int lane = threadIdx.x & 31;
  const int half = lane >> 4, m = lane & 15;
#pragma unroll
  for (int nt = 0; nt < 4; ++nt) {
    const int col = nt * 16 + m;
#pragma unroll
    for (int r = 0; r < 8; ++r) {
      const int row = rowBase + r + half * 8;
      if (row < nRows) out[(size_t)row * 64u + col] = fmaxf(acc[nt][r], 0.0f);
    }
  }
}

__global__ void __launch_bounds__(TPB) k_gemm_relu_mask_sum(
    const float* __restrict__ agg, const float* __restrict__ deg,
    const float* __restrict__ W, const int* __restrict__ keep,
    float* __restrict__ sumh, int nTiles, int nRows) {
  __shared__ float sWt[64 * WSTRIDE];
  __shared__ float ssum[64];
  load_Wt(W, sWt);
  if (threadIdx.x < 64) ssum[threadIdx.x] = 0.0f;
  __syncthreads();
  const int tile = blockIdx.x * (TPB / 32) + (threadIdx.x >> 5);
  const int lane = threadIdx.x & 31;
  const int half = lane >> 4, m = lane & 15;
  if (tile < nTiles) {
    v8f acc[4] = {};
    const int rowBase = tile * 16;
    gemm_tile(agg, deg, sWt, rowBase, nRows, acc);
#pragma unroll
    for (int nt = 0; nt < 4; ++nt) {
      float part = 0.0f;
#pragma unroll
      for (int r = 0; r < 8; ++r) {
        const int row = rowBase + r + half * 8;
        if (row < nRows) {
          float h = fmaxf(acc[nt][r], 0.0f);
          part += keep[row] ? h : 0.0f;
        }
      }
      atomicAdd(&ssum[nt * 16 + m], part);  // ds_add_f32
    }
  }
  __syncthreads();
  if (threadIdx.x < 64) {
    float s = ssum[threadIdx.x];
    if (s != 0.0f) atomicAdd(&sumh[threadIdx.x], s);
  }
}

// ---------------- BFS (exact synchronous rounds, integer hops) ----------------
// state: [0]=done [1]=cnt_prev [2]=hc [3]=cnt_iter [4..5]=argmin u64
//        [6]=hstar [7]=rneed [8]=lo [9]=hi [10]=tstar [11]=bsDone [12]=bsCnt
__global__ void k_fill_hop(int* __restrict__ hop, int n) {
  int i = blockIdx.x * blockDim.x + threadIdx.x;
  if (i < n) hop[i] = n;
}

__global__ void k_seed(const int* __restrict__ perm, int* __restrict__ hop, int* __restrict__ st) {
  hop[perm[0]] = 0;
  st[0] = 0;  // done
  st[1] = 1;  // cnt_prev (root reached)
  st[2] = 0;  // hc
}

__global__ void k_bfs_copy(const int* __restrict__ cur, int* __restrict__ nxt,
                           int* __restrict__ st, int n) {
  int i = blockIdx.x * blockDim.x + threadIdx.x;
  if (i < n) nxt[i] = cur[i];
  if (i == 0) {
    st[3] = 0;
    *(unsigned long long*)(st + 4) = ~0ULL;
  }
}

__global__ void k_bfs_relax(const int* __restrict__ cur, int* __restrict__ nxt,
                            const int* __restrict__ src, const int* __restrict__ dst,
                            const int* __restrict__ st, int nE, int n) {
  if (st[0]) return;
  int e = blockIdx.x * blockDim.x + threadIdx.x;
  if (e >= nE) return;
  int s = src[e], d = dst[e];
  int nh = cur[s] + (s != d);  // self-loop weight 0
  if (nh < n) atomicMin(&nxt[d], nh);
}

__global__ void k_bfs_count(const int* __restrict__ hop, const int* __restrict__ perm,
                            int* __restrict__ st, int n) {
  __shared__ int sc;
  __shared__ unsigned long long sa;
  if (st[0]) return;  // uniform across grid
  if (threadIdx.x == 0) { sc = 0; sa = ~0ULL; }
  __syncthreads();
  int i = blockIdx.x * blockDim.x + threadIdx.x;
  if (i < n) {
    if (hop[i] < n) {
      atomicAdd(&sc, 1);
    } else {  // unreached: argmin of perm (packed with index)
      unsigned long long key = ((unsigned long long)(unsigned)perm[i] << 32) | (unsigned)i;
      atomicMin(&sa, key);
    }
  }
  __syncthreads();
  if (threadIdx.x == 0) {
    if (sc) atomicAdd(&st[3], sc);
    if (sa != ~0ULL) atomicMin((unsigned long long*)(st + 4), sa);
  }
}

__global__ void k_bfs_step(int* __restrict__ st, int* __restrict__ hop, int target) {
  if (st[0]) return;
  st[2] += 1;  // hc++
  int cnt = st[3];
  if (cnt == st[1]) {  // stuck: seed unreached node with smallest perm at hop=hc
    unsigned long long am = *(unsigned long long*)(st + 4);
    if (am != ~0ULL) hop[(int)(am & 0xffffffffu)] = st[2];
    cnt += 1;
  }
  st[1] = cnt;
  if (cnt > target) st[0] = 1;
}

// ---------------- keep-mask selection ----------------
__global__ void k_hist(const int* __restrict__ hop, unsigned* __restrict__ hist, int n) {
  __shared__ unsigned sh[64];
  if (threadIdx.x < 64) sh[threadIdx.x] = 0;
  __syncthreads();
  int i = blockIdx.x * blockDim.x + threadIdx.x;
  if (i < n) {
    int h = hop[i];
    if ((unsigned)h < 64u) atomicAdd(&sh[h], 1u);
  }
  __syncthreads();
  if (threadIdx.x < 64 && sh[threadIdx.x]) atomicAdd(&hist[threadIdx.x], sh[threadIdx.x]);
}

__global__ void k_select(int* __restrict__ st, const unsigned* __restrict__ hist,
                         int ktarget, int n) {
  int cum = 0, hstar = 63, below = 0, found = 0;
  for (int h = 0; h < 64; ++h) {
    int c = (int)hist[h];
    if (!found && cum + c >= ktarget) { hstar = h; below = cum; found = 1; }
    cum += c;
  }
  if (!found) { hstar = 63; below = cum - (int)hist[63]; }
  int r = ktarget - below;
  st[6] = hstar; st[7] = r;
  st[8] = 0; st[9] = n;  // lo, hi
  st[12] = 0;            // count accumulator
  if (r <= 0) { st[10] = 0; st[11] = 1; }
  else        { st[10] = 0; st[11] = 0; }
}

__global__ void k_bs_count(const int* __restrict__ hop, const int* __restrict__ perm,
                           int* __restrict__ st, int n) {
  __shared__ int sc;
  if (st[11]) return;  // uniform
  if (threadIdx.x == 0) sc = 0;
  __syncthreads();
  int i = blockIdx.x * blockDim.x + threadIdx.x;
  int mid = (st[8] + st[9]) >> 1;
  if (i < n && hop[i] == st[6] && perm[i] < mid) atomicAdd(&sc, 1);
  __syncthreads();
  if (threadIdx.x == 0 && sc) atomicAdd(&st[12], sc);
}

__global__ void k_bs_step(int* __restrict__ st) {
  if (st[11]) return;
  int lo = st[8], hi = st[9];
  int mid = (lo + hi) >> 1;
  if (st[12] >= st[7]) hi = mid; else lo = mid;
  st[12] = 0;
  st[8] = lo; st[9] = hi;
  if (hi - lo <= 1) { st[10] = hi; st[11] = 1; }
}

__global__ void k_make_keep(const int* __restrict__ hop, const int* __restrict__ perm,
                            const int* __restrict__ st, int* __restrict__ keep, int n) {
  int i = blockIdx.x * blockDim.x + threadIdx.x;
  if (i >= n) return;
  int h = hop[i];
  keep[i] = (h < st[6]) || (h == st[6] && perm[i] < st[10]);
}

// ---------------- summary -> v, BCE, final ----------------
__global__ void k_summary_v(const float* __restrict__ sumh, const float* __restrict__ Wd,
                            float* __restrict__ v, float invk) {
  __shared__ float sm[64];
  int t = threadIdx.x;  // 64 threads
  float s = sumh[t] * invk;
  sm[t] = 1.0f / (1.0f + expf(-s));
  __syncthreads();
  const float* wr = Wd + t * 64;
  float acc = 0.0f;
#pragma unroll
  for (int c = 0; c < 64; ++c) acc += wr[c] * sm[c];
  v[t] = acc;
}

__global__ void k_bce(const float* __restrict__ pos, const float* __restrict__ neg,
                      const float* __restrict__ v, float* __restrict__ lossAcc, int n) {
  __shared__ float sl;
  if (threadIdx.x == 0) sl = 0.0f;
  __syncthreads();
  const int lane = threadIdx.x & 31;
  const int node = (blockIdx.x * blockDim.x + threadIdx.x) >> 5;
  if (node < n) {
    const float v0 = v[lane], v1 = v[lane + 32];
    const float* pr = pos + (size_t)node * 64u;
    const float* nr = neg + (size_t)node * 64u;
    float a = pr[lane] * v0 + pr[lane + 32] * v1;
    float b = nr[lane] * v0 + nr[lane + 32] * v1;
#pragma unroll
    for (int off = 16; off > 0; off >>= 1) {
      a += __shfl_xor(a, off, 32);
      b += __shfl_xor(b, off, 32);
    }
    if (lane == 0) atomicAdd(&sl, softplusf(-a) + softplusf(b));
  }
  __syncthreads();
  if (threadIdx.x == 0 && sl != 0.0f) atomicAdd(lossAcc, sl);
}

__global__ void k_final(const float* __restrict__ lossAcc, float* __restrict__ out, float invn) {
  out[0] = lossAcc[0] * invn;
}

// ---------------- launch ----------------
extern "C" void kernel_launch(void* const* d_in, const int* in_sizes, int n_in,
                              void* d_out, int out_size, void* d_ws, size_t ws_size,
                              hipStream_t stream) {
  (void)n_in; (void)out_size; (void)ws_size;
  const int N = in_sizes[0] / 64;
  const int E = in_sizes[3];
  const float* feat  = (const float*)d_in[0];
  const float* Wenc  = (const float*)d_in[1];
  const float* Wdisc = (const float*)d_in[2];
  const int* src = (const int*)d_in[3];
  const int* dst = (const int*)d_in[4];
  const int* perm_neg = (const int*)d_in[5];
  const int* perm1 = (const int*)d_in[6];
  const int* perm2 = (const int*)d_in[7];
  float* out = (float*)d_out;

  // workspace carve-up (256B aligned slices)
  char* w = (char*)d_ws;
  auto alloc = [&](size_t bytes) -> void* {
    void* p = (void*)w;
    w += (bytes + 255) & ~(size_t)255;
    return p;
  };
  float* pos  = (float*)alloc((size_t)N * 64 * 4);
  float* neg  = (float*)alloc((size_t)N * 64 * 4);
  float* agg  = (float*)alloc((size_t)N * 64 * 4);
  float* degA = (float*)alloc((size_t)N * 4);
  float* degM = (float*)alloc((size_t)N * 4);
  int* hopA   = (int*)alloc((size_t)N * 4);
  int* hopB   = (int*)alloc((size_t)N * 4);
  int* keep   = (int*)alloc((size_t)N * 4);
  unsigned* hist = (unsigned*)alloc(64 * 4);
  float* sumh = (float*)alloc(64 * 4);
  float* vvec = (float*)alloc(64 * 4);
  float* lossA = (float*)alloc(4);
  int* st = (int*)alloc(64 * 4);

  const int nodeBlocks = (N + TPB - 1) / TPB;
  const int edgeBlocks = (E + TPB - 1) / TPB;
  const int edgeWaveBlocks = (int)(((size_t)E * 32 + TPB - 1) / TPB);
  const int nTiles = (N + 15) / 16;
  const int gemmBlocks = (nTiles + (TPB / 32) - 1) / (TPB / 32);
  const int bceBlocks = (int)(((size_t)N * 32 + TPB - 1) / TPB);
  const int kTarget = (int)((double)N * 0.8);

  // degrees (all-ones edge mask)
  hipMemsetAsync(degA, 0, (size_t)N * 4, stream);
  k_deg<<<edgeBlocks, TPB, 0, stream>>>(dst, degA, E);

  // pos = GCN(feat)
  hipMemsetAsync(agg, 0, (size_t)N * 64 * 4, stream);
  k_agg<<<edgeWaveBlocks, TPB, 0, stream>>>(feat, src, dst, nullptr, agg, E);
  k_gemm_relu_store<<<gemmBlocks, TPB, 0, stream>>>(agg, degA, Wenc, pos, nTiles, N);

  // neg = GCN(feat[perm_neg])
  hipMemsetAsync(agg, 0, (size_t)N * 64 * 4, stream);
  k_agg<<<edgeWaveBlocks, TPB, 0, stream>>>(feat, src, dst, perm_neg, agg, E);
  k_gemm_relu_store<<<gemmBlocks, TPB, 0, stream>>>(agg, degA, Wenc, neg, nTiles, N);

  hipMemsetAsync(lossA, 0, 4, stream);

  const int* perms[2] = {perm1, perm2};
  for (int p = 0; p < 2; ++p) {
    const int* perm = perms[p];

    // ---- exact synchronous BFS with stall-seeding ----
    k_fill_hop<<<nodeBlocks, TPB, 0, stream>>>(hopA, N);
    k_seed<<<1, 1, 0, stream>>>(perm, hopA, st);
    for (int it = 0; it < MAX_BFS_ITERS; ++it) {
      int* cur = (it & 1) ? hopB : hopA;
      int* nxt = (it & 1) ? hopA : hopB;
      k_bfs_copy<<<nodeBlocks, TPB, 0, stream>>>(cur, nxt, st, N);
      k_bfs_relax<<<edgeBlocks, TPB, 0, stream>>>(cur, nxt, src, dst, st, E, N);
      k_bfs_count<<<nodeBlocks, TPB, 0, stream>>>(nxt, perm, st, N);
      k_bfs_step<<<1, 1, 0, stream>>>(st, nxt, kTarget);
    }
    int* hopF = (MAX_BFS_ITERS & 1) ? hopB : hopA;

    // ---- keep = 80% closest, ties by perm (hist + binary search) ----
    hipMemsetAsync(hist, 0, 64 * 4, stream);
    k_hist<<<nodeBlocks, TPB, 0, stream>>>(hopF, hist, N);
    k_select<<<1, 1, 0, stream>>>(st, hist, kTarget, N);
    for (int b = 0; b < BS_ITERS; ++b) {
      k_bs_count<<<nodeBlocks, TPB, 0, stream>>>(hopF, perm, st, N);
      k_bs_step<<<1, 1, 0, stream>>>(st);
    }
    k_make_keep<<<nodeBlocks, TPB, 0, stream>>>(hopF, perm, st, keep, N);

    // ---- masked GCN fused with summary accumulation ----
    hipMemsetAsync(agg, 0, (size_t)N * 64 * 4, stream);
    hipMemsetAsync(degM, 0, (size_t)N * 4, stream);
    hipMemsetAsync(sumh, 0, 64 * 4, stream);
    k_agg_masked<<<edgeWaveBlocks, TPB, 0, stream>>>(feat, src, dst, keep, agg, degM, E);
    k_gemm_relu_mask_sum<<<gemmBlocks, TPB, 0, stream>>>(agg, degM, Wenc, keep, sumh, nTiles, N);

    // ---- summary -> v, then BCE for this perm ----
    k_summary_v<<<1, 64, 0, stream>>>(sumh, Wdisc, vvec, 1.0f / (float)kTarget);
    k_bce<<<bceBlocks, TPB, 0, stream>>>(pos, neg, vvec, lossA, N);
  }
  k_final<<<1, 1, 0, stream>>>(lossA, out, 1.0f / (float)N);
}